// MambaModel_49486613185063
// MI455X (gfx1250) — compile-verified
//
#include <hip/hip_runtime.h>
#include <hip/hip_bf16.h>
#include <math.h>

typedef __attribute__((ext_vector_type(2))) float v2f;
typedef __attribute__((ext_vector_type(8))) float v8f;

#define BATCH 16
#define KPTS  512
#define LSEQ  1024
#define NDIMS 64
#define HD    512
#define NLAY  12
#define STATE 64
#define II    1024
#define DTR   32
#define CONVK 4
#define MTOK  (BATCH * LSEQ)    /* 16384 token rows */
#define SSMW  (DTR + 2 * STATE) /* 160 */

__device__ __forceinline__ float waveRedAdd(float v) {
#pragma unroll
  for (int off = 16; off > 0; off >>= 1) v += __shfl_xor(v, off, 32);
  return v;
}
__device__ __forceinline__ float fast_silu(float x) {
  return x / (1.f + __expf(-x));
}
__device__ __forceinline__ float fast_softplus(float x) {
  return (x > 20.f) ? x : log1pf(__expf(x));
}

// ---------------------------------------------------------------------------
// Token interleave: tokens[b, 2k, :] = xs[b,k,:]; tokens[b, 2k+1, 0] = ys[b,k]
// ---------------------------------------------------------------------------
__global__ __launch_bounds__(256) void token_kernel(const float* __restrict__ xs,
                                                    const float* __restrict__ ys,
                                                    float* __restrict__ tokens) {
  int t = blockIdx.x * 256 + threadIdx.x;       // < MTOK * NDIMS
  int d = t & (NDIMS - 1);
  int m = t >> 6;                               // b*LSEQ + l
  int l = m & (LSEQ - 1);
  int b = m >> 10;
  int kp = l >> 1;
  float v;
  if ((l & 1) == 0) v = xs[((size_t)b * KPTS + kp) * NDIMS + d];
  else              v = (d == 0) ? ys[(size_t)b * KPTS + kp] : 0.f;
  tokens[(size_t)t] = v;
}

// ---------------------------------------------------------------------------
// WMMA fp32 GEMM:  C[M,N] = A[M,K] * W[N,K]^T  (+ epilogue)
// One wave -> 64x32 tile: 4 A fragments x 2 B fragments -> 8 v8f accumulators.
// All shapes are template constants so k-offsets fold into the 24-bit
// immediate offset of global_load_b64 (no per-iteration 64-bit address math).
// Fragment loads software-pipelined one k-step ahead; grid sized so every
// wave owns a full tile (EXEC all-ones, required for WMMA).
// ---------------------------------------------------------------------------
enum GemmEpi { EPI_NONE = 0, EPI_BIAS = 1, EPI_SOFTPLUS = 2, EPI_ACCUM = 3 };

template <int EPI, int LDC>
__device__ __forceinline__ void store_acc8(v8f acc, float* __restrict__ C,
                                           int row0, int col, float bv) {
#pragma unroll
  for (int r = 0; r < 8; ++r) {
    float v = acc[r];
    float* p = C + (size_t)(row0 + r) * LDC + col;
    if (EPI == EPI_BIAS)     v += bv;
    if (EPI == EPI_SOFTPLUS) v = fast_softplus(v + bv);
    if (EPI == EPI_ACCUM)    v += *p;
    *p = v;
  }
}

template <int EPI, int LDA, int LDW, int LDC, int N, int K>
__global__ __launch_bounds__(256)
void gemm_tn_wmma(const float* __restrict__ A,
                  const float* __restrict__ W,
                  const float* __restrict__ bias,
                  float* __restrict__ C) {
  const int wave = (blockIdx.x << 3) + (threadIdx.x >> 5);
  const int lane = threadIdx.x & 31;
  constexpr int TILES_N = N / 32;               // 32-wide N tiles
  const int tm = wave / TILES_N;
  const int tn = wave - tm * TILES_N;
  const int m0 = tm << 6;
  const int n0 = tn << 5;
  const int lh = lane & 15;
  const int kk = (lane >> 4) << 1;              // K pair {0,1} / {2,3}

  v8f c00 = {}; v8f c01 = {}; v8f c10 = {}; v8f c11 = {};
  v8f c20 = {}; v8f c21 = {}; v8f c30 = {}; v8f c31 = {};

  const float* wp0 = W + (size_t)(n0 + lh) * LDW + kk;
  const float* wp1 = wp0 + (size_t)16 * LDW;
  const float* ap0 = A + (size_t)(m0 + lh) * LDA + kk;
  const float* ap1 = ap0 + (size_t)16 * LDA;
  const float* ap2 = ap0 + (size_t)32 * LDA;
  const float* ap3 = ap0 + (size_t)48 * LDA;

  // prologue: fragments for k = 0
  v2f b0 = *(const v2f*)(wp0);
  v2f b1 = *(const v2f*)(wp1);
  v2f a0 = *(const v2f*)(ap0);
  v2f a1 = *(const v2f*)(ap1);
  v2f a2 = *(const v2f*)(ap2);
  v2f a3 = *(const v2f*)(ap3);

#pragma unroll 8
  for (int k = 4; k <= K; k += 4) {
    // prefetch next k-step's fragments (clamped on final iteration)
    const int kn = (k < K) ? k : 0;
    v2f nb0 = *(const v2f*)(wp0 + kn);
    v2f nb1 = *(const v2f*)(wp1 + kn);
    v2f na0 = *(const v2f*)(ap0 + kn);
    v2f na1 = *(const v2f*)(ap1 + kn);
    v2f na2 = *(const v2f*)(ap2 + kn);
    v2f na3 = *(const v2f*)(ap3 + kn);
    if (k + 32 <= K) {                          // pull upcoming lines to WGP
      __builtin_prefetch(wp0 + k + 32, 0, 3);
      __builtin_prefetch(ap0 + k + 32, 0, 3);
    }
    c00 = __builtin_amdgcn_wmma_f32_16x16x4_f32(false, a0, false, b0, 0, c00, false, false);
    c01 = __builtin_amdgcn_wmma_f32_16x16x4_f32(false, a0, false, b1, 0, c01, false, false);
    c10 = __builtin_amdgcn_wmma_f32_16x16x4_f32(false, a1, false, b0, 0, c10, false, false);
    c11 = __builtin_amdgcn_wmma_f32_16x16x4_f32(false, a1, false, b1, 0, c11, false, false);
    c20 = __builtin_amdgcn_wmma_f32_16x16x4_f32(false, a2, false, b0, 0, c20, false, false);
    c21 = __builtin_amdgcn_wmma_f32_16x16x4_f32(false, a2, false, b1, 0, c21, false, false);
    c30 = __builtin_amdgcn_wmma_f32_16x16x4_f32(false, a3, false, b0, 0, c30, false, false);
    c31 = __builtin_amdgcn_wmma_f32_16x16x4_f32(false, a3, false, b1, 0, c31, false, false);
    b0 = nb0; b1 = nb1; a0 = na0; a1 = na1; a2 = na2; a3 = na3;
  }

  float bv0 = 0.f, bv1 = 0.f;
  if (EPI == EPI_BIAS || EPI == EPI_SOFTPLUS) {
    bv0 = bias[n0 + lh];
    bv1 = bias[n0 + 16 + lh];
  }
  const int rbase = (lane >> 4) << 3;           // lanes 16-31 hold M=r+8
  const int col0 = n0 + lh;
  const int col1 = n0 + 16 + lh;
  store_acc8<EPI, LDC>(c00, C, m0 +  0 + rbase, col0, bv0);
  store_acc8<EPI, LDC>(c01, C, m0 +  0 + rbase, col1, bv1);
  store_acc8<EPI, LDC>(c10, C, m0 + 16 + rbase, col0, bv0);
  store_acc8<EPI, LDC>(c11, C, m0 + 16 + rbase, col1, bv1);
  store_acc8<EPI, LDC>(c20, C, m0 + 32 + rbase, col0, bv0);
  store_acc8<EPI, LDC>(c21, C, m0 + 32 + rbase, col1, bv1);
  store_acc8<EPI, LDC>(c30, C, m0 + 48 + rbase, col0, bv0);
  store_acc8<EPI, LDC>(c31, C, m0 + 48 + rbase, col1, bv1);
}

// ---------------------------------------------------------------------------
// RMSNorm: one wave per row of 512
// ---------------------------------------------------------------------------
__global__ __launch_bounds__(256) void rmsnorm_kernel(const float* __restrict__ X,
                                                      const float* __restrict__ w,
                                                      float* __restrict__ Y) {
  int row = (blockIdx.x << 3) + (threadIdx.x >> 5);
  int lane = threadIdx.x & 31;
  const float* xr = X + (size_t)row * HD;
  float ss = 0.f;
#pragma unroll
  for (int j = lane; j < HD; j += 32) { float v = xr[j]; ss += v * v; }
  ss = waveRedAdd(ss);
  float scale = rsqrtf(ss * (1.f / HD) + 1e-5f);
  float* yr = Y + (size_t)row * HD;
#pragma unroll
  for (int j = lane; j < HD; j += 32) yr[j] = xr[j] * scale * w[j];
}

// ---------------------------------------------------------------------------
// Causal depthwise conv1d (K=4) + bias + silu.  u0 lives in proj cols [0,II).
// ---------------------------------------------------------------------------
__global__ __launch_bounds__(256) void conv_kernel(const float* __restrict__ proj,
                                                   const float* __restrict__ cw,
                                                   const float* __restrict__ cb,
                                                   float* __restrict__ u) {
  int t = blockIdx.x * 256 + threadIdx.x;       // = (b*LSEQ + l)*II + i
  int i = t & (II - 1);
  int l = (t >> 10) & (LSEQ - 1);
  int b = t >> 20;
  size_t rowbase = (size_t)b * LSEQ * (2 * II);
  float acc = cb[i];
#pragma unroll
  for (int k = 0; k < CONVK; ++k) {
    int ls = l - (CONVK - 1) + k;
    if (ls >= 0) acc += proj[rowbase + (size_t)ls * (2 * II) + i] * cw[i * CONVK + k];
  }
  u[(size_t)t] = fast_silu(acc);
}

// ---------------------------------------------------------------------------
// Selective scan: one wave per (b, i); lane owns states n=lane and n=lane+32.
// y = sum_n s_n * C_n via wave shfl-xor reduction; gated output overwrites
// proj cols [0,II) (the spent u0 slots), gate read from cols [II,2I).
// ---------------------------------------------------------------------------
__global__ __launch_bounds__(256) void scan_kernel(const float* __restrict__ dtb,
                                                   const float* __restrict__ u,
                                                   float* proj,
                                                   const float* __restrict__ ssm,
                                                   const float* __restrict__ Alog,
                                                   const float* __restrict__ Dp) {
  int wave = (blockIdx.x << 3) + (threadIdx.x >> 5);  // = b*II + i
  int lane = threadIdx.x & 31;
  int b = wave >> 10;
  int i = wave & (II - 1);
  float A0 = -__expf(Alog[(size_t)i * STATE + lane]);
  float A1 = -__expf(Alog[(size_t)i * STATE + lane + 32]);
  float Dv = Dp[i];
  float s0 = 0.f, s1 = 0.f;
  size_t mb = (size_t)b * LSEQ;
  for (int l = 0; l < LSEQ; ++l) {
    size_t m = mb + l;
    float dtv = dtb[m * II + i];
    float uv  = u[m * II + i];
    const float* sr = ssm + m * SSMW;
    float Bv0 = sr[DTR + lane];
    float Bv1 = sr[DTR + 32 + lane];
    float Cv0 = sr[DTR + STATE + lane];
    float Cv1 = sr[DTR + STATE + 32 + lane];
    float du = dtv * uv;
    s0 = __expf(dtv * A0) * s0 + du * Bv0;
    s1 = __expf(dtv * A1) * s1 + du * Bv1;
    float y = s0 * Cv0 + s1 * Cv1;
    y = waveRedAdd(y);
    if (lane == 0) {
      float g = proj[m * (2 * II) + II + i];
      proj[m * (2 * II) + i] = (y + uv * Dv) * fast_silu(g);
    }
  }
}

// ---------------------------------------------------------------------------
// Final rmsnorm + readout at even token positions: one wave per (b, k).
// ---------------------------------------------------------------------------
__global__ __launch_bounds__(256) void final_kernel(const float* __restrict__ h,
                                                    const float* __restrict__ nw,
                                                    const float* __restrict__ rw,
                                                    const float* __restrict__ rb,
                                                    float* __restrict__ out) {
  int wave = (blockIdx.x << 3) + (threadIdx.x >> 5);  // = b*KPTS + k
  int lane = threadIdx.x & 31;
  int b = wave >> 9;
  int kp = wave & (KPTS - 1);
  const float* xr = h + ((size_t)b * LSEQ + 2 * kp) * HD;
  float ss = 0.f, dp = 0.f;
#pragma unroll
  for (int j = lane; j < HD; j += 32) {
    float v = xr[j];
    ss += v * v;
    dp += v * nw[j] * rw[j];
  }
  ss = waveRedAdd(ss);
  dp = waveRedAdd(dp);
  if (lane == 0) {
    float scale = rsqrtf(ss * (1.f / HD) + 1e-5f);
    out[wave] = dp * scale + rb[0];
  }
}

// ---------------------------------------------------------------------------
extern "C" void kernel_launch(void* const* d_in, const int* in_sizes, int n_in,
                              void* d_out, int out_size, void* d_ws, size_t ws_size,
                              hipStream_t stream) {
  (void)in_sizes; (void)n_in; (void)out_size; (void)ws_size;
  const float* xs         = (const float*)d_in[0];
  const float* ys         = (const float*)d_in[1];
  const float* read_in_w  = (const float*)d_in[2];
  const float* read_in_b  = (const float*)d_in[3];
  const float* norm_w     = (const float*)d_in[4];
  const float* in_proj_w  = (const float*)d_in[5];
  const float* conv_w     = (const float*)d_in[6];
  const float* conv_b     = (const float*)d_in[7];
  const float* x_proj_w   = (const float*)d_in[8];
  const float* dt_proj_w  = (const float*)d_in[9];
  const float* dt_proj_b  = (const float*)d_in[10];
  const float* A_log      = (const float*)d_in[11];
  const float* Dp         = (const float*)d_in[12];
  const float* out_proj_w = (const float*)d_in[13];
  const float* norm_f_w   = (const float*)d_in[14];
  const float* read_out_w = (const float*)d_in[15];
  const float* read_out_b = (const float*)d_in[16];
  float* out = (float*)d_out;

  float* ws = (float*)d_ws;
  size_t off = 0;
  float* tokens = ws + off; off += (size_t)MTOK * NDIMS;     // 1 M floats
  float* h      = ws + off; off += (size_t)MTOK * HD;        // 8 M
  float* xn     = ws + off; off += (size_t)MTOK * HD;        // 8 M
  float* proj   = ws + off; off += (size_t)MTOK * 2 * II;    // 32 M
  float* u      = ws + off; off += (size_t)MTOK * II;        // 16 M
  float* ssm    = ws + off; off += (size_t)MTOK * SSMW;      // 2.5 M
  float* dtb    = ws + off; off += (size_t)MTOK * II;        // 16 M

  // Grids: tiles = (MTOK/64) * (N/32), 8 waves per block -> tiles/8 blocks.
  token_kernel<<<(MTOK * NDIMS) / 256, 256, 0, stream>>>(xs, ys, tokens);
  gemm_tn_wmma<EPI_BIAS, NDIMS, NDIMS, HD, HD, NDIMS>
      <<<(256 * (HD / 32)) / 8, 256, 0, stream>>>(tokens, read_in_w, read_in_b, h);

  for (int layer = 0; layer < NLAY; ++layer) {
    rmsnorm_kernel<<<MTOK / 8, 256, 0, stream>>>(h, norm_w + (size_t)layer * HD, xn);
    gemm_tn_wmma<EPI_NONE, HD, HD, 2 * II, 2 * II, HD>
        <<<(256 * (2 * II / 32)) / 8, 256, 0, stream>>>(
            xn, in_proj_w + (size_t)layer * 2 * II * HD, nullptr, proj);
    conv_kernel<<<(MTOK * II) / 256, 256, 0, stream>>>(
        proj, conv_w + (size_t)layer * II * CONVK, conv_b + (size_t)layer * II, u);
    gemm_tn_wmma<EPI_NONE, II, II, SSMW, SSMW, II>
        <<<(256 * (SSMW / 32)) / 8, 256, 0, stream>>>(
            u, x_proj_w + (size_t)layer * SSMW * II, nullptr, ssm);
    gemm_tn_wmma<EPI_SOFTPLUS, SSMW, DTR, II, II, DTR>
        <<<(256 * (II / 32)) / 8, 256, 0, stream>>>(
            ssm, dt_proj_w + (size_t)layer * II * DTR,
            dt_proj_b + (size_t)layer * II, dtb);
    scan_kernel<<<MTOK / 8, 256, 0, stream>>>(
        dtb, u, proj, ssm, A_log + (size_t)layer * II * STATE,
        Dp + (size_t)layer * II);
    gemm_tn_wmma<EPI_ACCUM, 2 * II, II, HD, HD, II>
        <<<(256 * (HD / 32)) / 8, 256, 0, stream>>>(
            proj, out_proj_w + (size_t)layer * HD * II, nullptr, h);
  }

  final_kernel<<<(BATCH * KPTS) / 8, 256, 0, stream>>>(
      h, norm_f_w, read_out_w, read_out_b, out);
}